// SpikeLayer_75617194213630
// MI455X (gfx1250) — compile-verified
//
#include <hip/hip_runtime.h>

// Integrate-and-fire spiking layer for MI455X (gfx1250, wave32).
//
// V_t = V_{t-1} + x_t;  V_t = 0 where |V_t| > 1;  emit V_t for t=0..15.
// x: [16, 1024, 4096] f32. Pure streaming: 512 MB traffic, ~0.2 GFLOP ->
// HBM-bound floor ~22 us at 23.3 TB/s. No matrix structure -> no WMMA.
//
// This version maximizes per-wave memory-level parallelism: phase 1 issues
// all 16 independent global_load_b128 (NT) into a live register array so the
// backend can emit them as one clause with staggered s_wait_loadcnt; phase 2
// runs the register-carried recurrence and streams NT stores. All addresses
// are 32-bit offsets in GVS (saddr + voffset, scale_offset) form.

typedef float v4f __attribute__((ext_vector_type(4)));

#define SPIKE_TIMES 16
#define SPIKE_THRESH 1.0f

__global__ __launch_bounds__(256) void spike_if_kernel(
    const float* __restrict__ in, float* __restrict__ out, unsigned int n4) {
  // n4 = float4s per time step (chunk * hidden / 4).
  const unsigned int gid = blockIdx.x * blockDim.x + threadIdx.x;
  if (gid >= n4) return;

  const v4f* __restrict__ pin = (const v4f*)in;
  v4f* __restrict__ pout = (v4f*)out;

  // ---- Phase 1: issue all 16 independent streaming loads (batch / clause) ----
  v4f I[SPIKE_TIMES];
#pragma unroll
  for (int t = 0; t < SPIKE_TIMES; ++t) {
    const unsigned int off = gid + (unsigned int)t * n4;  // fits in 32 bits
    I[t] = __builtin_nontemporal_load(pin + off);
  }

  // ---- Phase 2: register-carried recurrence + streaming stores ----
  v4f V;
  V.x = 0.0f; V.y = 0.0f; V.z = 0.0f; V.w = 0.0f;

#pragma unroll
  for (int t = 0; t < SPIKE_TIMES; ++t) {
    V.x += I[t].x; V.y += I[t].y; V.z += I[t].z; V.w += I[t].w;
    V.x = (__builtin_fabsf(V.x) > SPIKE_THRESH) ? 0.0f : V.x;
    V.y = (__builtin_fabsf(V.y) > SPIKE_THRESH) ? 0.0f : V.y;
    V.z = (__builtin_fabsf(V.z) > SPIKE_THRESH) ? 0.0f : V.z;
    V.w = (__builtin_fabsf(V.w) > SPIKE_THRESH) ? 0.0f : V.w;
    const unsigned int off = gid + (unsigned int)t * n4;
    __builtin_nontemporal_store(V, pout + off);
  }
}

extern "C" void kernel_launch(void* const* d_in, const int* in_sizes, int n_in,
                              void* d_out, int out_size, void* d_ws, size_t ws_size,
                              hipStream_t stream) {
  (void)n_in; (void)out_size; (void)d_ws; (void)ws_size;

  const float* in = (const float*)d_in[0];
  float* out = (float*)d_out;

  const unsigned int total = (unsigned int)in_sizes[0];   // 16*1024*4096
  const unsigned int step_elems = total / SPIKE_TIMES;    // 4,194,304
  const unsigned int n4 = step_elems >> 2;                // 1,048,576

  const int block = 256;                                  // 8 wave32 waves
  const int grid = (int)((n4 + block - 1) / block);       // 4096 blocks

  spike_if_kernel<<<grid, block, 0, stream>>>(in, out, n4);
}